// FCond_13589276525159
// MI455X (gfx1250) — compile-verified
//
#include <hip/hip_runtime.h>
#include <math.h>

typedef float v2f __attribute__((ext_vector_type(2)));
typedef float v8f __attribute__((ext_vector_type(8)));

#if __has_builtin(__builtin_amdgcn_tanhf)
#define TANHF(v) __builtin_amdgcn_tanhf(v)
#else
#define TANHF(v) tanhf(v)
#endif

#define BATCH 32
#define NPTS  100000
#define CD    128
// threads per batch in main kernel: NPTS/4
#define TPB_BATCH 25000

// ---------------- conditioning GEMM via V_WMMA_F32_16X16X4_F32 ----------------
// Computes scale = sigmoid(c @ Ws^T + bs) and bias = c @ Wb^T + bb
// for t in {0,1,2}.  Output layout in ws: scale[3][32][3] at 0, bias[3][32][3] at 288.
struct CondParams {
  const float* c;        // [32][128]
  const float* Wm[6];    // index = t*2 + kind (kind 0 = Ws, 1 = Wb), each [3][128]
  const float* bvv[6];   // bs / bb, each [3]
  float* out;            // ws base
};

__global__ void cond_wmma_kernel(CondParams P) {
  const int t    = blockIdx.x >> 1;
  const int kind = blockIdx.x & 1;
  const float* W  = P.Wm[blockIdx.x];
  const float* bv = P.bvv[blockIdx.x];

  const int lane = threadIdx.x & 31;
  const int wave = threadIdx.x >> 5;
  const int m0   = wave << 4;          // M tile: rows m0..m0+15 of c
  const int row  = lane & 15;          // A: M index ; B: N index
  const int kh   = (lane >> 4) << 1;   // lane half -> K offset 0 or 2
  const int ncl  = (row < 3) ? row : 0;
  const bool nvalid = (row < 3);

  const float* crow = P.c + (m0 + row) * CD;
  const float* wrow = W + ncl * CD;

  v8f acc = {0.f,0.f,0.f,0.f,0.f,0.f,0.f,0.f};
  #pragma unroll 4
  for (int k = 0; k < CD; k += 4) {
    v2f a, b;
    a.x = crow[k + kh];
    a.y = crow[k + kh + 1];
    float w0 = wrow[k + kh];
    float w1 = wrow[k + kh + 1];
    b.x = nvalid ? w0 : 0.0f;
    b.y = nvalid ? w1 : 0.0f;
    acc = __builtin_amdgcn_wmma_f32_16x16x4_f32(
        /*neg_a=*/false, a, /*neg_b=*/false, b,
        /*c_mod=*/(short)0, acc, /*reuse_a=*/false, /*reuse_b=*/false);
  }

  const float bval = bv[ncl];
  float* outp = P.out + kind * 288 + t * 96;   // [32][3]
  const int mhi = (lane >> 4) << 3;            // C/D layout: lanes>=16 hold M=r+8
  if (nvalid) {
    #pragma unroll
    for (int r = 0; r < 8; ++r) {
      int m = m0 + r + mhi;
      float val = acc[r] + bval;
      if (kind == 0) val = 1.0f / (1.0f + __expf(-val));   // sigmoid for scale
      outp[m * 3 + row] = val;
    }
  }
}

// ---------------- fold: Meff = W^T*scale (row-wise), veff = b*scale + bias ----
struct FoldParams {
  const float* W[3];     // [3][3]
  const float* bvec[3];  // [3]
  const float* raw;      // ws: scale at 0, bias at 288
  float* cons;           // ws+576: [6 stages][32 b][12]
};

__global__ void fold_kernel(FoldParams P) {
  int idx = blockIdx.x * blockDim.x + threadIdx.x;   // 0..2303
  int stage = idx / 384;
  int rem   = idx - stage * 384;
  int b     = rem / 12;
  int j     = rem - b * 12;
  int t     = (stage < 2) ? stage : 2;               // stages 2..5 share tag 2
  int d     = (j < 9) ? (j / 3) : (j - 9);
  float s   = P.raw[t * 96 + b * 3 + d];
  float val;
  if (j < 9) val = P.W[t][j] * s;
  else       val = P.bvec[t][d] * s + P.raw[288 + t * 96 + b * 3 + d];
  P.cons[idx] = val;
}

// ---------------- fused streaming kernel: 6 stages in registers -------------
__global__ void __launch_bounds__(256) film_main_kernel(
    const float* __restrict__ x, const float* __restrict__ cons,
    float* __restrict__ out) {
  __shared__ float lc[144];                 // 2 batch slots x 6 stages x 12
  const int tid = blockIdx.x * 256 + threadIdx.x;
  const int bf  = (blockIdx.x * 256) / TPB_BATCH;   // first batch this block touches
  const int lt  = threadIdx.x;
  if (lt < 72) {
    int s = lt / 12, j = lt - s * 12;
    lc[lt] = cons[s * 384 + bf * 12 + j];
    int b2 = bf + 1;
    lc[72 + lt] = (b2 < BATCH) ? cons[s * 384 + b2 * 12 + j] : 0.0f;
  }
  __syncthreads();

  const int b = tid / TPB_BATCH;
  const float* Cs = &lc[(b - bf) * 72];

  const float4* X4 = (const float4*)x;
  float4 va = X4[tid * 3 + 0];
  float4 vb = X4[tid * 3 + 1];
  float4 vc = X4[tid * 3 + 2];
  float px[4] = {va.x, va.w, vb.z, vc.y};
  float py[4] = {va.y, vb.x, vb.w, vc.z};
  float pz[4] = {va.z, vb.y, vc.x, vc.w};

  #pragma unroll
  for (int s = 0; s < 6; ++s) {
    const float* Cp = Cs + s * 12;
    float m00=Cp[0],m01=Cp[1],m02=Cp[2],m10=Cp[3],m11=Cp[4],m12=Cp[5],
          m20=Cp[6],m21=Cp[7],m22=Cp[8],v0=Cp[9],v1=Cp[10],v2=Cp[11];
    #pragma unroll
    for (int p = 0; p < 4; ++p) {
      float nx = TANHF(fmaf(m02, pz[p], fmaf(m01, py[p], fmaf(m00, px[p], v0))));
      float ny = TANHF(fmaf(m12, pz[p], fmaf(m11, py[p], fmaf(m10, px[p], v1))));
      float nz = TANHF(fmaf(m22, pz[p], fmaf(m21, py[p], fmaf(m20, px[p], v2))));
      px[p] = nx; py[p] = ny; pz[p] = nz;
    }
  }

  float4* O4 = (float4*)out;
  O4[tid * 3 + 0] = make_float4(px[0], py[0], pz[0], px[1]);
  O4[tid * 3 + 1] = make_float4(py[1], pz[1], px[2], py[2]);
  O4[tid * 3 + 2] = make_float4(pz[2], px[3], py[3], pz[3]);
}

extern "C" void kernel_launch(void* const* d_in, const int* in_sizes, int n_in,
                              void* d_out, int out_size, void* d_ws, size_t ws_size,
                              hipStream_t stream) {
  // d_in: 0=t 1=x 2=c then per tag {W,b,Ws,bs,Wb,bb} starting at 3
  float* ws = (float*)d_ws;

  CondParams cp;
  cp.c = (const float*)d_in[2];
  for (int t = 0; t < 3; ++t) {
    int base = 3 + t * 6;
    cp.Wm [t * 2 + 0] = (const float*)d_in[base + 2];  // Ws
    cp.bvv[t * 2 + 0] = (const float*)d_in[base + 3];  // bs
    cp.Wm [t * 2 + 1] = (const float*)d_in[base + 4];  // Wb
    cp.bvv[t * 2 + 1] = (const float*)d_in[base + 5];  // bb
  }
  cp.out = ws;
  cond_wmma_kernel<<<6, 64, 0, stream>>>(cp);

  FoldParams fp;
  for (int t = 0; t < 3; ++t) {
    fp.W[t]    = (const float*)d_in[3 + t * 6];
    fp.bvec[t] = (const float*)d_in[3 + t * 6 + 1];
  }
  fp.raw  = ws;
  fp.cons = ws + 576;
  fold_kernel<<<9, 256, 0, stream>>>(fp);

  // 32*100000/4 points-per-thread-group = 800000 threads = 3125 blocks of 256
  film_main_kernel<<<3125, 256, 0, stream>>>((const float*)d_in[1], ws + 576,
                                             (float*)d_out);
}